// _ScaledDotProductAttention_16088947491565
// MI455X (gfx1250) — compile-verified
//
#include <hip/hip_runtime.h>
#include <math.h>

// Residual scaled-dot-product attention, fully fused per 16-row block.
// MI455X (gfx1250): wave32, V_WMMA_F32_16X16X4_F32 for exact-f32 matmuls,
// 320KB LDS/WGP lets us hold a full 16x2048 f32 score block (128KB) so the
// 512MB scores/attn arrays touch HBM exactly once each (write-only, NT).
// HBM floor: prev(512MB R) + scores(512MB W) + attn(512MB W) + ~66MB => ~70us.

#define BSZ   2
#define NH    16
#define SEQ   2048
#define DKD   64
#define ROWS  16           // WMMA M
#define WAVES 8
#define NTHREADS (WAVES * 32)

typedef float v2f __attribute__((ext_vector_type(2)));
typedef float v4f __attribute__((ext_vector_type(4)));
typedef float v8f __attribute__((ext_vector_type(8)));

__global__ __launch_bounds__(NTHREADS)
void attn_fused_kernel(const float* __restrict__ q,      // [BS,NH,SEQ,DK]
                       const float* __restrict__ k,      // [BS,NH,DK,SEQ] (pre-transposed)
                       const float* __restrict__ v,      // [BS,NH,SEQ,DK]
                       const float* __restrict__ prev,   // [BS,NH,SEQ,SEQ]
                       const float* __restrict__ scale,  // [1]
                       const unsigned char* __restrict__ kpm, // [BS,SEQ] bool
                       float* __restrict__ out,          // [BS,NH,SEQ,DK]
                       float* __restrict__ attn,         // [BS,NH,SEQ,SEQ]
                       float* __restrict__ scores)       // [BS,NH,SEQ,SEQ]
{
    __shared__ float lds_s[ROWS * SEQ];   // score block, then attn block in-place (128KB)
    __shared__ float lds_o[ROWS * DKD];   // out partials, ds_add_f32 accumulated (4KB)

    const int tid   = threadIdx.x;
    const int wave  = tid >> 5;
    const int lane  = tid & 31;
    const int lhalf = lane >> 4;          // 0 | 1 (lane group)
    const int l16   = lane & 15;

    const int bh   = blockIdx.y;          // 0..31
    const int b    = bh >> 4;
    const int row0 = blockIdx.x * ROWS;

    const float* qb = q    + (size_t)bh * SEQ * DKD;
    const float* kb = k    + (size_t)bh * DKD * SEQ;
    const float* vb = v    + (size_t)bh * SEQ * DKD;
    const float* pb = prev + (size_t)bh * SEQ * SEQ;
    float*       sb = scores + (size_t)bh * SEQ * SEQ;
    float*       ab = attn   + (size_t)bh * SEQ * SEQ;
    float*       ob = out    + (size_t)bh * SEQ * DKD;
    const unsigned char* mb = kpm + (size_t)b * SEQ;

    const float scl = scale[0];

    // zero the out-partial accumulation buffer (untouched until phase 3)
    for (int e = tid; e < ROWS * DKD; e += NTHREADS) lds_o[e] = 0.0f;

    // ---------------- Phase 1: scores = (q @ k)*scale + prev, masked ----------------
    // A fragments (16x4 f32): lane holds row m=l16, K = 4*kk + 2*lhalf + {0,1}.
    v2f afrag[16];
    {
        const float* arow = qb + (size_t)(row0 + l16) * DKD + 2 * lhalf;
        #pragma unroll
        for (int kk = 0; kk < 16; ++kk)
            afrag[kk] = *(const v2f*)(arow + kk * 4);
    }

    #pragma unroll 1
    for (int it = 0; it < 16; ++it) {                // 16 N-tiles per wave, contiguous
        const int n0  = (wave * 16 + it) * 16;
        const int col = n0 + l16;
        const bool masked = (mb[col] != 0);

        v8f acc = {};
        #pragma unroll
        for (int kk = 0; kk < 16; ++kk) {            // K = 64 in steps of 4
            const int kd = kk * 4 + 2 * lhalf;
            v2f bfrag;
            bfrag.x = kb[(size_t)kd       * SEQ + col];
            bfrag.y = kb[(size_t)(kd + 1) * SEQ + col];
            acc = __builtin_amdgcn_wmma_f32_16x16x4_f32(
                      false, afrag[kk], false, bfrag, (short)0, acc, false, false);
        }

        // Unconditional prev loads (batched, NT) -- mask applied via select,
        // NOT via control flow, so the 512MB prev stream is never predicated.
        float pv[8];
        #pragma unroll
        for (int i = 0; i < 8; ++i) {
            const int m = i + 8 * lhalf;
            pv[i] = __builtin_nontemporal_load(pb + (size_t)(row0 + m) * SEQ + col);
        }
        #pragma unroll
        for (int i = 0; i < 8; ++i) {                // C/D: m = i + 8*lhalf, n = col
            const int m = i + 8 * lhalf;
            float s = acc[i] * scl + pv[i];
            s = masked ? -INFINITY : s;              // v_cndmask, no branch
            lds_s[m * SEQ + col] = s;
            __builtin_nontemporal_store(s, sb + (size_t)(row0 + m) * SEQ + col);
        }
    }

    __syncthreads();

    // ---------------- Phase 2: softmax over each row (2 rows per wave) ---------------
    // float4 accesses: 16 b128 LDS ops per pass instead of 64 b32.
    #pragma unroll 1
    for (int rr = 0; rr < 2; ++rr) {
        const int m = wave * 2 + rr;
        v4f* rowv = (v4f*)(lds_s + m * SEQ);         // 512 float4, 16B aligned

        float mx = -INFINITY;
        #pragma unroll 1
        for (int c = lane; c < SEQ / 4; c += 32) {
            v4f x = rowv[c];
            mx = fmaxf(mx, fmaxf(fmaxf(x.x, x.y), fmaxf(x.z, x.w)));
        }
        #pragma unroll
        for (int o = 16; o >= 1; o >>= 1) mx = fmaxf(mx, __shfl_xor(mx, o, 32));

        float sum = 0.0f;
        #pragma unroll 1
        for (int c = lane; c < SEQ / 4; c += 32) {
            v4f x = rowv[c];
            v4f e;
            e.x = __expf(x.x - mx);                  // exp(-inf - mx) = 0 for masked
            e.y = __expf(x.y - mx);
            e.z = __expf(x.z - mx);
            e.w = __expf(x.w - mx);
            rowv[c] = e;
            sum += (e.x + e.y) + (e.z + e.w);
        }
        #pragma unroll
        for (int o = 16; o >= 1; o >>= 1) sum += __shfl_xor(sum, o, 32);

        const float inv = 1.0f / sum;
        v4f* arowv = (v4f*)(ab + (size_t)(row0 + m) * SEQ);
        #pragma unroll 1
        for (int c = lane; c < SEQ / 4; c += 32) {
            v4f a = rowv[c];
            a.x *= inv; a.y *= inv; a.z *= inv; a.w *= inv;
            rowv[c] = a;                             // attn kept in LDS for phase 3
            __builtin_nontemporal_store(a, arowv + c);   // b128 NT store
        }
    }

    __syncthreads();

    // ---------------- Phase 3: out = attn @ v, K=2048 split across 8 waves -----------
    v8f oacc[4];
    #pragma unroll
    for (int t = 0; t < 4; ++t) oacc[t] = (v8f){};

    const int kbase = wave * (SEQ / WAVES);          // 256 K-values per wave
    #pragma unroll 1
    for (int ks = 0; ks < (SEQ / WAVES) / 4; ++ks) { // 64 steps of K=4
        const int kk0 = kbase + ks * 4 + 2 * lhalf;
        v2f a = *(const v2f*)(lds_s + l16 * SEQ + kk0);   // A from LDS attn block
        #pragma unroll
        for (int t = 0; t < 4; ++t) {                // N = 64 in 4 tiles, A reused
            const int n = t * 16 + l16;
            v2f bfrag;
            bfrag.x = vb[(size_t)kk0       * DKD + n];
            bfrag.y = vb[(size_t)(kk0 + 1) * DKD + n];
            oacc[t] = __builtin_amdgcn_wmma_f32_16x16x4_f32(
                          false, a, false, bfrag, (short)0, oacc[t], false, false);
        }
    }
    // cross-wave reduction through LDS float atomics (ds_add_f32)
    #pragma unroll
    for (int t = 0; t < 4; ++t) {
        #pragma unroll
        for (int i = 0; i < 8; ++i) {
            const int m = i + 8 * lhalf;
            const int n = t * 16 + l16;
            atomicAdd(&lds_o[m * DKD + n], oacc[t][i]);
        }
    }

    __syncthreads();

    for (int e = tid; e < ROWS * DKD; e += NTHREADS) {
        const int m = e / DKD, n = e % DKD;
        ob[(size_t)(row0 + m) * DKD + n] = lds_o[e];
    }
}

extern "C" void kernel_launch(void* const* d_in, const int* in_sizes, int n_in,
                              void* d_out, int out_size, void* d_ws, size_t ws_size,
                              hipStream_t stream) {
    const float*         q    = (const float*)d_in[0];
    const float*         k    = (const float*)d_in[1];
    const float*         v    = (const float*)d_in[2];
    const float*         prev = (const float*)d_in[3];
    const float*         scl  = (const float*)d_in[4];
    const unsigned char* kpm  = (const unsigned char*)d_in[5];

    float* out    = (float*)d_out;                                  // [2,16,2048,64]
    float* attn   = out  + (size_t)BSZ * NH * SEQ * DKD;            // [2,16,2048,2048]
    float* scores = attn + (size_t)BSZ * NH * SEQ * SEQ;            // [2,16,2048,2048]

    dim3 grid(SEQ / ROWS, BSZ * NH);   // 128 x 32 workgroups
    attn_fused_kernel<<<grid, NTHREADS, 0, stream>>>(q, k, v, prev, scl, kpm,
                                                     out, attn, scores);
}